// Dense_RBS_state_vector_3298534883716
// MI455X (gfx1250) — compile-verified
//
#include <hip/hip_runtime.h>
#include <cstdint>
#include <cmath>

// Problem constants (match reference)
#define NQ     128                 // qubits
#define DIM    8128                // C(128,2)
#define BATCH  8192
#define D4     (DIM / 4)           // 2032 float4 columns per row
#define GY     256                 // row groups
#define ROWS   (BATCH / GY)        // 32 rows per block

typedef float v4f __attribute__((ext_vector_type(4)));

// ---------------------------------------------------------------------------
// Kernel 1: total[d] = prod_g cos(angle_g)^mask[g,d], computed analytically.
// For pair (a,b), a<b: gates with exactly one of {a,b} in {g,g+1} are
//   b>a+1 : {a-1, a, b-1, b} ∩ [0,126]
//   b==a+1: {a-1, a+1}       ∩ [0,126]
// ---------------------------------------------------------------------------
__global__ __launch_bounds__(256)
void rbs_scale_kernel(const float* __restrict__ angles, float* __restrict__ total) {
    int d = blockIdx.x * blockDim.x + threadIdx.x;
    if (d >= DIM) return;

    // invert row-major triu enumeration: off(a) = a*(255-a)/2
    int a = (int)((255.0 - sqrt(65025.0 - 8.0 * (double)d)) * 0.5);
    if (a < 0) a = 0;
    while (((a + 1) * (255 - (a + 1))) / 2 <= d) ++a;
    while ((a * (255 - a)) / 2 > d) --a;
    int b = a + 1 + (d - (a * (255 - a)) / 2);

    float t = 1.0f;
    if (b == a + 1) {
        if (a >= 1)   t *= cosf(angles[a - 1]);
        if (b <= 126) t *= cosf(angles[b]);
    } else {
        if (a >= 1)   t *= cosf(angles[a - 1]);
        t *= cosf(angles[a]);
        t *= cosf(angles[b - 1]);
        if (b <= 126) t *= cosf(angles[b]);
    }
    total[d] = t;
}

// ---------------------------------------------------------------------------
// Kernel 2: out = in * total (broadcast over rows). Pure HBM stream.
// Column-persistent scale: staged via CDNA5 async global->LDS DMA, then held
// in registers for 32 rows of NT b128 load / mul / NT b128 store.
// ---------------------------------------------------------------------------
__global__ __launch_bounds__(256)
void rbs_apply_kernel(const float* __restrict__ in,
                      const float* __restrict__ scale,
                      float* __restrict__ out) {
    __shared__ v4f s_scale[256];

    const int tid  = threadIdx.x;
    const int col4 = blockIdx.x * 256 + tid;     // float4 column index
    const bool active = (col4 < D4);

    // Stage this block's 4KB scale slice into LDS with the async engine
    // (ASYNCcnt-tracked; per-lane EXEC masks OOB lanes).
    if (active) {
        unsigned lds_addr = (unsigned)(uintptr_t)(&s_scale[tid]);
        uint64_t gaddr    = (uint64_t)(uintptr_t)(scale + (size_t)col4 * 4);
        asm volatile("global_load_async_to_lds_b128 %0, %1, off"
                     :
                     : "v"(lds_addr), "v"(gaddr)
                     : "memory");
    }
    asm volatile("s_wait_asynccnt 0" ::: "memory");
    __syncthreads();
    if (!active) return;

    const v4f s = s_scale[tid];                  // persistent in VGPRs

    const size_t base = (size_t)blockIdx.y * ROWS * (size_t)D4 + (size_t)col4;
    const v4f* __restrict__ ip = (const v4f*)in + base;
    v4f* __restrict__       op = (v4f*)out + base;

    // 532 MB total stream: mark both directions non-temporal so the stream
    // doesn't evict the resident 32KB scale vector / other L2 state.
#pragma unroll 4
    for (int r = 0; r < ROWS; ++r) {
        v4f v = __builtin_nontemporal_load(ip);
        v.x *= s.x; v.y *= s.y; v.z *= s.z; v.w *= s.w;
        __builtin_nontemporal_store(v, op);
        ip += D4;
        op += D4;
    }
}

// ---------------------------------------------------------------------------
extern "C" void kernel_launch(void* const* d_in, const int* in_sizes, int n_in,
                              void* d_out, int out_size, void* d_ws, size_t ws_size,
                              hipStream_t stream) {
    const float* input  = (const float*)d_in[0];   // [BATCH, DIM] f32
    const float* angles = (const float*)d_in[1];   // [127] f32
    // d_in[2] = gate_masks: unused (masks derived analytically)
    float* out   = (float*)d_out;                  // [BATCH, DIM] f32
    float* total = (float*)d_ws;                   // [DIM] f32 scratch

    rbs_scale_kernel<<<(DIM + 255) / 256, 256, 0, stream>>>(angles, total);

    dim3 grid((D4 + 255) / 256, GY);               // (8, 256)
    rbs_apply_kernel<<<grid, 256, 0, stream>>>(input, total, out);
}